// HERGAST_5944234737752
// MI455X (gfx1250) — compile-verified
//
#include <hip/hip_runtime.h>
#include <math.h>

typedef float v2f __attribute__((ext_vector_type(2)));
typedef float v8f __attribute__((ext_vector_type(8)));

#define NEG_SLOPE 0.2f
#define EPS_DEN   1e-16f

#define AP 36   // As pitch: 16B-aligned rows, conflict-free frag reads
#define BP 72   // Bs pitch: row-delta-2 -> +16 banks, halves disjoint; 16B-aligned

// =====================================================================
// GEMM: C[M,Nn] = act(A[M,K] @ B[K,Nn] + bias)   (fp32, WMMA 16x16x4)
// Block = 256 thr = 8 waves; block tile = 64(M) x 64(N).
// Each wave: 32(M) x 16(N) = two 16x16 accumulators sharing one B frag.
// Interior blocks take a uniform-branch float4 tile-load fast path;
// edge blocks use scalar zero-padded loads. Inner loop branch-free.
// grid.z selects relation slice of B and C. Requires K % 4 == 0.
// =====================================================================
__global__ __launch_bounds__(256)
void gemm_f32_wmma(const float* __restrict__ A, const float* __restrict__ B,
                   const float* __restrict__ bias, float* __restrict__ C,
                   int M, int K, int Nn,
                   long long bStrideZ, long long cStrideZ, int act)
{
  B += (long long)blockIdx.z * bStrideZ;
  C += (long long)blockIdx.z * cStrideZ;

  __shared__ float As[64 * AP];   // 64 rows x 32 k
  __shared__ float Bs[32 * BP];   // 32 k   x 64 cols

  const int m0   = blockIdx.x * 64;
  const int nb0  = blockIdx.y * 64;
  const int wave = threadIdx.x >> 5;
  const int lane = threadIdx.x & 31;
  const int l16  = lane & 15;
  const int half = lane >> 4;                // 0: K=ks,ks+1  1: K=ks+2,ks+3
  const int mrow = (wave & 1) * 32;
  const int ncol = (wave >> 1) * 16;

  const bool interior = (m0 + 64 <= M) && (nb0 + 64 <= Nn);  // block-uniform

  v8f acc0 = {0.f,0.f,0.f,0.f,0.f,0.f,0.f,0.f};
  v8f acc1 = {0.f,0.f,0.f,0.f,0.f,0.f,0.f,0.f};

  for (int k0 = 0; k0 < K; k0 += 32) {
    if (interior && (k0 + 32 <= K)) {
      // ---- fast path: unconditional float4 loads (uniform branch) ----
      #pragma unroll
      for (int j = 0; j < 2; ++j) {                    // A: 512 float4
        int f = threadIdx.x + j * 256;
        int r = f >> 3, c = (f & 7) << 2;              // 8 f4 per 32-f row
        float4 v = *(const float4*)&A[(long long)(m0 + r) * K + k0 + c];
        *(float4*)&As[r * AP + c] = v;
      }
      #pragma unroll
      for (int j = 0; j < 2; ++j) {                    // B: 512 float4
        int f = threadIdx.x + j * 256;
        int r = f >> 4, c = (f & 15) << 2;             // 16 f4 per 64-f row
        float4 v = *(const float4*)&B[(long long)(k0 + r) * Nn + nb0 + c];
        *(float4*)&Bs[r * BP + c] = v;
      }
    } else {
      // ---- slow path: scalar zero-padded loads ----
      #pragma unroll
      for (int j = 0; j < 8; ++j) {
        int i = threadIdx.x + j * 256;
        int r = i >> 5, c = i & 31;
        int gm = m0 + r, gk = k0 + c;
        As[r * AP + c] = (gm < M && gk < K) ? A[(long long)gm * K + gk] : 0.f;
      }
      #pragma unroll
      for (int j = 0; j < 8; ++j) {
        int i = threadIdx.x + j * 256;
        int r = i >> 6, c = i & 63;
        int gk = k0 + r, gn = nb0 + c;
        Bs[r * BP + c] = (gk < K && gn < Nn) ? B[(long long)gk * Nn + gn] : 0.f;
      }
    }
    __syncthreads();

    // ---- 8 k-steps of 4, branch-free, 2 WMMAs each ----
    #pragma unroll
    for (int ks = 0; ks < 32; ks += 4) {
      const int ka = ks + (half << 1);
      v2f b, a0, a1;
      b.x  = Bs[ka * BP + ncol + l16];
      b.y  = Bs[(ka + 1) * BP + ncol + l16];
      a0.x = As[(mrow + l16) * AP + ka];
      a0.y = As[(mrow + l16) * AP + ka + 1];
      a1.x = As[(mrow + 16 + l16) * AP + ka];
      a1.y = As[(mrow + 16 + l16) * AP + ka + 1];
      acc0 = __builtin_amdgcn_wmma_f32_16x16x4_f32(false, a0, false, b,
                                                   (short)0, acc0, false, false);
      acc1 = __builtin_amdgcn_wmma_f32_16x16x4_f32(false, a1, false, b,
                                                   (short)0, acc1, false, false);
    }
    __syncthreads();
  }

  // ---- epilogue: C/D layout VGPR j -> rows (base+j, base+j+8) ----
  const int col = nb0 + ncol + l16;
  if (col < Nn) {
    const float bv = bias ? bias[col] : 0.f;
    const int r0 = m0 + mrow + (half << 3);
    #pragma unroll
    for (int j = 0; j < 8; ++j) {
      int row = r0 + j;
      if (row < M) {
        float v = acc0[j] + bv;
        if (act == 1) v = (v > 0.f) ? v : expm1f(v);
        C[(long long)row * Nn + col] = v;
      }
    }
    #pragma unroll
    for (int j = 0; j < 8; ++j) {
      int row = r0 + 16 + j;
      if (row < M) {
        float v = acc1[j] + bv;
        if (act == 1) v = (v > 0.f) ? v : expm1f(v);
        C[(long long)row * Nn + col] = v;
      }
    }
  }
}

// ---------------------------------------------------------------------
// qn[r,n] = xw[r,n,:] . q ;  kn[r,n] = xw[r,n,:] . k   (one wave per row)
// ---------------------------------------------------------------------
__global__ __launch_bounds__(256)
void qk_dots(const float* __restrict__ xw, const float* __restrict__ q,
             const float* __restrict__ k, float* __restrict__ qn,
             float* __restrict__ kn, int N, int C, int R)
{
  const long long w = (long long)blockIdx.x * (blockDim.x >> 5) + (threadIdx.x >> 5);
  const int lane = threadIdx.x & 31;
  const long long total = (long long)R * N;
  if (w >= total) return;
  const float* row = xw + w * C;
  float sq = 0.f, sk = 0.f;
  for (int c = lane; c < C; c += 32) {
    float v = row[c];
    sq += v * q[c];
    sk += v * k[c];
  }
  for (int off = 16; off > 0; off >>= 1) {
    sq += __shfl_down(sq, off, 32);
    sk += __shfl_down(sk, off, 32);
  }
  if (lane == 0) { qn[w] = sq; kn[w] = sk; }
}

// ---------------------------------------------------------------------
// fills (grid-stride)
// ---------------------------------------------------------------------
__global__ void fill_f32(float* p, float v, long long n) {
  long long i = (long long)blockIdx.x * blockDim.x + threadIdx.x;
  long long step = (long long)gridDim.x * blockDim.x;
  for (; i < n; i += step) p[i] = v;
}
__global__ void fill_u32(unsigned* p, unsigned v, long long n) {
  long long i = (long long)blockIdx.x * blockDim.x + threadIdx.x;
  long long step = (long long)gridDim.x * blockDim.x;
  for (; i < n; i += step) p[i] = v;
}

// ---------------------------------------------------------------------
// Edge pass 1: alpha = leaky_relu(qn[et,dst] + kn[et,src]); segment max
// via monotonic-uint atomicMax (sign-flip key).
// ---------------------------------------------------------------------
__global__ __launch_bounds__(256)
void edge_alpha_max(const int* __restrict__ src, const int* __restrict__ dst,
                    const int* __restrict__ et, const float* __restrict__ qn,
                    const float* __restrict__ kn, float* __restrict__ alpha,
                    unsigned* __restrict__ maxkey, int E, int N)
{
  int e = blockIdx.x * blockDim.x + threadIdx.x;
  if (e >= E) return;
  int s = src[e], d = dst[e], r = et[e];
  float a = qn[r * N + d] + kn[r * N + s];
  a = (a > 0.f) ? a : NEG_SLOPE * a;
  alpha[e] = a;
  unsigned u = __float_as_uint(a);
  unsigned key = (u & 0x80000000u) ? ~u : (u | 0x80000000u);
  atomicMax(&maxkey[d], key);
}

// ---------------------------------------------------------------------
// Edge pass 2: ea = exp(alpha - m[dst]); denom[dst] += ea (in place)
// ---------------------------------------------------------------------
__global__ __launch_bounds__(256)
void edge_expsum(const int* __restrict__ dst, float* __restrict__ alpha,
                 const unsigned* __restrict__ maxkey, float* __restrict__ denom,
                 int E)
{
  int e = blockIdx.x * blockDim.x + threadIdx.x;
  if (e >= E) return;
  int d = dst[e];
  unsigned key = maxkey[d];
  unsigned u = (key & 0x80000000u) ? (key & 0x7FFFFFFFu) : ~key;
  float m = __uint_as_float(u);
  float ea = expf(alpha[e] - m);
  alpha[e] = ea;
  atomicAdd(&denom[d], ea);
}

// ---------------------------------------------------------------------
// Edge pass 3: out[dst,:] += (ea/(denom[dst]+eps)) * xw[et,src,:]
// One wave per edge: coalesced row gather, L2 atomic scatter.
// ---------------------------------------------------------------------
__global__ __launch_bounds__(256)
void edge_scatter(const int* __restrict__ src, const int* __restrict__ dst,
                  const int* __restrict__ et, const float* __restrict__ ea,
                  const float* __restrict__ denom, const float* __restrict__ xw,
                  float* __restrict__ out, int E, int N, int C)
{
  int wv = blockIdx.x * (blockDim.x >> 5) + (threadIdx.x >> 5);
  int lane = threadIdx.x & 31;
  if (wv >= E) return;
  int s = src[wv], d = dst[wv], r = et[wv];
  float w = ea[wv] / (denom[d] + EPS_DEN);
  const float* xr = xw + ((long long)r * N + s) * C;
  float* orow = out + (long long)d * C;
  for (int c = lane; c < C; c += 32)
    atomicAdd(&orow[c], w * xr[c]);
}

// ---------------------------------------------------------------------
// ELU in place
// ---------------------------------------------------------------------
__global__ void elu_inplace(float* p, long long n) {
  long long i = (long long)blockIdx.x * blockDim.x + threadIdx.x;
  long long step = (long long)gridDim.x * blockDim.x;
  for (; i < n; i += step) {
    float v = p[i];
    p[i] = (v > 0.f) ? v : expm1f(v);
  }
}

// =====================================================================
extern "C" void kernel_launch(void* const* d_in, const int* in_sizes, int n_in,
                              void* d_out, int out_size, void* d_ws, size_t ws_size,
                              hipStream_t stream)
{
  (void)in_sizes; (void)n_in; (void)out_size; (void)ws_size;
  const int N = 50000, E = 800000, IN = 200, HID = 256, OUT = 64, R = 2;

  const float* feat  = (const float*)d_in[0];
  const int*   eidx  = (const int*)d_in[1];
  const int*   etype = (const int*)d_in[2];
  const float* w1    = (const float*)d_in[3];
  const float* q1    = (const float*)d_in[4];
  const float* k1    = (const float*)d_in[5];
  const float* w2    = (const float*)d_in[6];
  const float* q2    = (const float*)d_in[7];
  const float* k2    = (const float*)d_in[8];
  const float* dW1   = (const float*)d_in[9];
  const float* db1   = (const float*)d_in[10];
  const float* dW2   = (const float*)d_in[11];
  const float* db2   = (const float*)d_in[12];

  const int* srcI = eidx;       // edge_index[0]
  const int* dstI = eidx + E;   // edge_index[1]

  float* out = (float*)d_out;
  float* h2  = out;                          // [N, OUT]
  float* h3  = out + (size_t)N * OUT;        // [N, IN]

  float* ws = (float*)d_ws;
  size_t o = 0;
  float*    xw1    = ws + o; o += (size_t)R * N * HID;   // 25.6M f
  float*    aggr1  = ws + o; o += (size_t)N * HID;       // 12.8M f
  float*    xw2    = ws + o; o += (size_t)R * N * OUT;   // 6.4M f
  float*    qn     = ws + o; o += (size_t)R * N;
  float*    kn     = ws + o; o += (size_t)R * N;
  unsigned* maxkey = (unsigned*)(ws + o); o += N;
  float*    denom  = ws + o; o += N;
  float*    alpha  = ws + o; o += E;
  float*    mid    = xw1;   // decoder hidden reuses dead xw1 region

  const dim3 blk(256);

  // ---------------- Layer 1 ----------------
  {
    dim3 g((N + 63) / 64, (HID + 63) / 64, R);
    gemm_f32_wmma<<<g, blk, 0, stream>>>(feat, w1, nullptr, xw1, N, IN, HID,
                                         (long long)IN * HID, (long long)N * HID, 0);
  }
  {
    long long waves = (long long)R * N;
    qk_dots<<<dim3((unsigned)((waves + 7) / 8)), blk, 0, stream>>>(
        xw1, q1, k1, qn, kn, N, HID, R);
  }
  fill_u32<<<dim3((N + 255) / 256), blk, 0, stream>>>(maxkey, 0u, N);
  fill_f32<<<dim3((N + 255) / 256), blk, 0, stream>>>(denom, 0.f, N);
  fill_f32<<<dim3(4096), blk, 0, stream>>>(aggr1, 0.f, (long long)N * HID);
  edge_alpha_max<<<dim3((E + 255) / 256), blk, 0, stream>>>(
      srcI, dstI, etype, qn, kn, alpha, maxkey, E, N);
  edge_expsum<<<dim3((E + 255) / 256), blk, 0, stream>>>(dstI, alpha, maxkey, denom, E);
  edge_scatter<<<dim3((E + 7) / 8), blk, 0, stream>>>(
      srcI, dstI, etype, alpha, denom, xw1, aggr1, E, N, HID);
  elu_inplace<<<dim3(4096), blk, 0, stream>>>(aggr1, (long long)N * HID);  // h1

  // ---------------- Layer 2 ----------------
  {
    dim3 g((N + 63) / 64, (OUT + 63) / 64, R);
    gemm_f32_wmma<<<g, blk, 0, stream>>>(aggr1, w2, nullptr, xw2, N, HID, OUT,
                                         (long long)HID * OUT, (long long)N * OUT, 0);
  }
  {
    long long waves = (long long)R * N;
    qk_dots<<<dim3((unsigned)((waves + 7) / 8)), blk, 0, stream>>>(
        xw2, q2, k2, qn, kn, N, OUT, R);
  }
  fill_u32<<<dim3((N + 255) / 256), blk, 0, stream>>>(maxkey, 0u, N);
  fill_f32<<<dim3((N + 255) / 256), blk, 0, stream>>>(denom, 0.f, N);
  fill_f32<<<dim3(4096), blk, 0, stream>>>(h2, 0.f, (long long)N * OUT);
  edge_alpha_max<<<dim3((E + 255) / 256), blk, 0, stream>>>(
      srcI, dstI, etype, qn, kn, alpha, maxkey, E, N);
  edge_expsum<<<dim3((E + 255) / 256), blk, 0, stream>>>(dstI, alpha, maxkey, denom, E);
  edge_scatter<<<dim3((E + 7) / 8), blk, 0, stream>>>(
      srcI, dstI, etype, alpha, denom, xw2, h2, E, N, OUT);
  elu_inplace<<<dim3(4096), blk, 0, stream>>>(h2, (long long)N * OUT);

  // ---------------- Decoder ----------------
  {
    dim3 g((N + 63) / 64, (HID + 63) / 64, 1);
    gemm_f32_wmma<<<g, blk, 0, stream>>>(h2, dW1, db1, mid, N, OUT, HID, 0, 0, 0);
  }
  {
    dim3 g((N + 63) / 64, (IN + 63) / 64, 1);
    gemm_f32_wmma<<<g, blk, 0, stream>>>(mid, dW2, db2, h3, N, HID, IN, 0, 0, 0);
  }
}